// ICON_Transformer_20091857011279
// MI455X (gfx1250) — compile-verified
//
#include <hip/hip_runtime.h>
#include <math.h>

typedef float v2f __attribute__((ext_vector_type(2)));
typedef float v8f __attribute__((ext_vector_type(8)));

#define NHB 7
#define TT  8
#define EE  16
#define WAVES 8                    // waves per block
#define PPW  4                     // points per wave
#define PTS_PER_BLOCK (WAVES * PPW)

__device__ __forceinline__ unsigned sel4(unsigned a0, unsigned a1, unsigned a2,
                                         unsigned a3, int q) {
    const unsigned lo = (q & 1) ? a1 : a0;
    const unsigned hi = (q & 1) ? a3 : a2;
    return (q & 2) ? hi : lo;
}

__device__ __forceinline__ void dist_angle(float lon1, float lat1, float lon2, float lat2,
                                           float& d, float& phi) {
    float dlon = lon2 - lon1;
    float sl1 = __sinf(lat1), cl1 = __cosf(lat1);
    float sl2 = __sinf(lat2), cl2 = __cosf(lat2);
    float cdl = __cosf(dlon), sdl = __sinf(dlon);
    float cosd = fmaf(cl1 * cl2, cdl, sl1 * sl2);
    cosd = fminf(fmaxf(cosd, -1.0f + 1e-6f), 1.0f - 1e-6f);
    d = acosf(cosd);
    phi = atan2f(sdl * cl2, cl1 * sl2 - sl1 * cl2 * cdl);
}

__global__ void __launch_bounds__(256)
icon_attn_wmma(const float* __restrict__ x,          // (B,N,E)
               const float* __restrict__ coords,     // (2,N)
               const float* __restrict__ cref,       // (2,B,N)
               const float* __restrict__ ccross,     // (2,B,N,T)
               const float* __restrict__ sigma_d,    // (1,)
               const float* __restrict__ kappa_vm,   // (1,)
               const int*   __restrict__ adjc,       // (N,NH)
               const unsigned char* __restrict__ amask, // (N,NH) bool
               float* __restrict__ out,              // (B,N,T,E)
               int Bn, int Nn)
{
    // Per-wave LDS: two 16x16 W tiles (stride 17 -> conflict-free column
    // reads), neighbor dist/angle, neighbor indices for 4 points.
    __shared__ float sW  [WAVES][2][16][17];
    __shared__ float sDn [WAVES][PPW][8];
    __shared__ float sPn [WAVES][PPW][8];
    __shared__ int   sIdx[WAVES][PPW][8];

    const int wave = threadIdx.x >> 5;
    const int lane = threadIdx.x & 31;

    const int P  = Bn * Nn;                          // 655,368 fits in int
    const int pb = (blockIdx.x * WAVES + wave) * PPW;

    // Per-point wave-uniform info (registers, constant-indexable in phase 2).
    unsigned nq_[PPW], bnq_[PPW], xof_[PPW], vld_[PPW];
    #pragma unroll
    for (int q = 0; q < PPW; ++q) {
        const int p  = pb + q;
        const int v  = (p < P);
        const int cp = v ? p : (P - 1);
        const int b  = cp / Nn, n = cp - b * Nn;
        nq_[q]  = (unsigned)n;
        bnq_[q] = (unsigned)b * Nn + n;              // (b,n) flat index
        xof_[q] = (unsigned)b * (unsigned)(Nn * EE); // x batch base (elements)
        vld_[q] = (unsigned)v;
    }
    const unsigned latOf = (unsigned)Bn * Nn;        // cref channel stride
    const unsigned crOf  = latOf * TT;               // ccross channel stride

    const float sigma = sigma_d[0];
    const float kappa = kappa_vm[0];
    const float expk  = __expf(kappa);

    // ---------- Phase 1a: 32 lanes = 4 points x 8 slots (h==7 is pad) ------
    {
        const int q = lane >> 3;
        const int h = lane & 7;
        const unsigned nq  = sel4(nq_[0],  nq_[1],  nq_[2],  nq_[3],  q);
        const unsigned bnq = sel4(bnq_[0], bnq_[1], bnq_[2], bnq_[3], q);
        if (h < NHB) {
            const int idx = adjc[nq * NHB + h];
            const float lonn = coords[(unsigned)idx];
            const float latn = coords[(unsigned)(Nn + idx)];
            const float rlon = cref[bnq];
            const float rlat = cref[latOf + bnq];
            float d, ph;
            dist_angle(rlon, rlat, lonn, latn, d, ph);
            if (amask[nq * NHB + h]) d = 1e10f;
            sDn[wave][q][h]  = d;
            sPn[wave][q][h]  = ph;
            sIdx[wave][q][h] = idx;
        } else {
            sDn[wave][q][h]  = 0.f;
            sPn[wave][q][h]  = 0.f;
            sIdx[wave][q][h] = 0;       // pad slot: valid index, weight is 0
        }
    }
    __syncthreads();

    // ---------- Phase 1b: 32 lanes = 4 points x 8 t-rows (fully occupied) --
    {
        const int q = lane >> 3;
        const int t = lane & 7;
        const unsigned bnq = sel4(bnq_[0], bnq_[1], bnq_[2], bnq_[3], q);
        const float rlon = cref[bnq];
        const float rlat = cref[latOf + bnq];
        const float clon = ccross[bnq * TT + t];
        const float clat = ccross[crOf + bnq * TT + t];
        float dt, pt;
        dist_angle(rlon, rlat, clon, clat, dt, pt);

        float s[NHB];
        float mx = -1e30f;
        #pragma unroll
        for (int h = 0; h < NHB; ++h) {
            const float dn = sDn[wave][q][h];
            const float pn = sPn[wave][q][h];
            float vm = __expf(kappa * __cosf(pt - pn));
            const float z  = (dt - dn) / sigma;
            const float nd = __expf(-0.5f * z * z);
            if (nd == 1.0f) vm = expk;       // reference's exact-match rule
            const float l = vm * nd;
            s[h] = l;
            mx = fmaxf(mx, l);
        }
        float Z = 0.f;
        #pragma unroll
        for (int h = 0; h < NHB; ++h) { s[h] = __expf(s[h] - mx); Z += s[h]; }
        const float inv = 1.0f / Z;

        // Block-diagonal A within tile (q>>1): even point -> cols 0..6,
        // odd point -> cols 8..14; cols 7 and 15 stay zero for every row, so
        // pad rows of B are annihilated by the MAC (no guards needed later).
        const int tile  = q >> 1;
        const int row   = (q & 1) * 8 + t;
        const int cbase = (q & 1) * 8;
        #pragma unroll
        for (int c = 0; c < 16; ++c) {
            const int rel = c - cbase;
            float v = 0.f;
            if (rel >= 0 && rel < NHB) v = s[rel] * inv;
            sW[wave][tile][row][c] = v;
        }
    }
    __syncthreads();

    // ---------- Phase 2: 2 tiles x 4x V_WMMA_F32_16X16X4_F32 (K = 16) ------
    // A 16x4 layout: lanes 0-15 hold M=lane, K=k0..k0+1; lanes 16-31 hold
    //                M=lane-16, K=k0+2..k0+3.
    // B 4x16 layout: vgpr0 lanes 0-15 = row k0, lanes 16-31 = row k0+2;
    //                vgpr1 = rows k0+1 / k0+3.  (C-style row striping)
    // Rows k0..k0+3 of a k-step all lie in one 8-row half, so the source
    // point index qrow is compile-time -> constant-indexed base offsets.
    const unsigned e = lane & 15;
    const int hiSel  = lane >> 4;         // 0: lanes 0-15, 1: lanes 16-31
    const int m      = lane & 15;

    #pragma unroll
    for (int tile = 0; tile < 2; ++tile) {
        v8f acc = {};
        #pragma unroll
        for (int s4 = 0; s4 < 4; ++s4) {
            const int k0   = s4 * 4 + hiSel * 2;
            const int qrow = tile * 2 + (s4 >> 1);     // compile-time
            const unsigned xo = xof_[qrow];

            v2f a;
            a.x = sW[wave][tile][m][k0];
            a.y = sW[wave][tile][m][k0 + 1];

            v2f bm;
            {
                const int hh = k0 & 7;
                const unsigned idx = (unsigned)sIdx[wave][qrow][hh];
                bm.x = x[xo + idx * EE + e];
            }
            {
                const int hh = (k0 + 1) & 7;
                const unsigned idx = (unsigned)sIdx[wave][qrow][hh];
                bm.y = x[xo + idx * EE + e];
            }

            // D = A*B + C, full f32. EXEC is all-ones here.
            acc = __builtin_amdgcn_wmma_f32_16x16x4_f32(
                false, a, false, bm, (short)0, acc, false, false);
        }

        // Store: D vgpr v = rows v (lanes<16 -> even point) / v+8 (odd point).
        // Each store instruction covers two contiguous 64B segments.
        {
            const unsigned valid = hiSel ? vld_[tile * 2 + 1] : vld_[tile * 2];
            const unsigned bnq   = hiSel ? bnq_[tile * 2 + 1] : bnq_[tile * 2];
            if (valid) {
                float* __restrict__ op = out + (size_t)bnq * (TT * EE);
                #pragma unroll
                for (int t = 0; t < TT; ++t)
                    op[(unsigned)(t * EE) + e] = acc[t];
            }
        }
    }
}

extern "C" void kernel_launch(void* const* d_in, const int* in_sizes, int n_in,
                              void* d_out, int out_size, void* d_ws, size_t ws_size,
                              hipStream_t stream) {
    (void)n_in; (void)out_size; (void)d_ws; (void)ws_size;
    const float* x      = (const float*)d_in[0];
    const float* coords = (const float*)d_in[1];
    const float* cref   = (const float*)d_in[2];
    const float* ccross = (const float*)d_in[3];
    const float* sigma  = (const float*)d_in[4];
    const float* kappa  = (const float*)d_in[5];
    const int*   adjc   = (const int*)d_in[6];
    const unsigned char* amask = (const unsigned char*)d_in[7];
    float* out = (float*)d_out;

    const int N = in_sizes[1] / 2;                 // coords is (2,N)
    const int B = in_sizes[0] / (N * EE);          // x is (B,N,E)
    const long P = (long)B * N;
    const int blocks = (int)((P + PTS_PER_BLOCK - 1) / PTS_PER_BLOCK);

    icon_attn_wmma<<<dim3(blocks), dim3(256), 0, stream>>>(
        x, coords, cref, ccross, sigma, kappa, adjc, amask, out, B, N);
}